// mpNet_72438918414851
// MI455X (gfx1250) — compile-verified
//
#include <hip/hip_runtime.h>
#include <hip/hip_bf16.h>

// ---------------------------------------------------------------------------
// MPNet matching pursuit, restructured to avoid materializing M_D (512MB).
//   corr[b,a] = sum_n conj(W[n,a]) * u[b,n],  u[b,n] = sum_m M[b,n,m] r[b,m]
// Only column norms md2[b,a] = sum_m |M_D[b,m,a]|^2 are precomputed (16MB).
// All GEMM work uses V_WMMA_F32_16X16X4_F32 (fp32 matrix path, wave32).
// ---------------------------------------------------------------------------

#define B_SZ   1024
#define N_ANT  64
#define M_MEAS 16
#define D_A    4096
#define ITERS  20

typedef float v2f __attribute__((ext_vector_type(2)));
typedef float v8f __attribute__((ext_vector_type(8)));

__device__ __forceinline__ v8f wmma_f32(v2f a, v2f b, v8f c) {
  // D = A(16x4) * B(4x16) + C(16x16), fp32.
  // 8 args: (neg_a, A, neg_b, B, c_mod, C, reuse_a, reuse_b).
  // NEG[1:0] must be 0 for f32 per ISA -> never negate A/B here.
  return __builtin_amdgcn_wmma_f32_16x16x4_f32(false, a, false, b, (short)0, c,
                                               false, false);
}

// ---------------------------------------------------------------------------
// e_norm_inv[a] = 1 / ||W[:,a]||
// ---------------------------------------------------------------------------
__global__ void enorm_kernel(const float* __restrict__ Wre,
                             const float* __restrict__ Wim,
                             float* __restrict__ enorm_inv) {
  int a = blockIdx.x * blockDim.x + threadIdx.x;
  if (a < D_A) {
    float s = 0.f;
    for (int n = 0; n < N_ANT; ++n) {
      float r = Wre[n * D_A + a], i = Wim[n * D_A + a];
      s += r * r + i * i;
    }
    enorm_inv[a] = 1.0f / sqrtf(s);
  }
}

// ---------------------------------------------------------------------------
// md2inv[b,a] = 1 / sum_m |M_D[b,m,a]|^2
// M_D[m,a] = sum_n conj(M[b,n,m]) * W[n,a]
//   re = Mr^T Wr + Mi^T Wi ; im = Mr^T Wi - Mi^T Wr
// One block per b (8 waves); each wave owns a-tiles (strided).
// ---------------------------------------------------------------------------
__global__ void __launch_bounds__(256)
norms_kernel(const float* __restrict__ Mre, const float* __restrict__ Mim,
             const float* __restrict__ Wre, const float* __restrict__ Wim,
             float* __restrict__ md2inv) {
  const int b = blockIdx.x;
  const int tid = threadIdx.x;
  const int lane = tid & 31;
  const int wave = tid >> 5;

  __shared__ float sMre[N_ANT * M_MEAS];
  __shared__ float sMim[N_ANT * M_MEAS];
  for (int i = tid; i < N_ANT * M_MEAS; i += 256) {
    sMre[i] = Mre[(size_t)b * (N_ANT * M_MEAS) + i];
    sMim[i] = Mim[(size_t)b * (N_ANT * M_MEAS) + i];
  }
  __syncthreads();

  // A-matrix (16x4 per k-step): row m = lane&15, k = (lane>>4)*2 + v.
  const int mrow  = lane & 15;
  const int khalf = (lane >> 4) << 1;
  v2f ar[16], ai[16];
#pragma unroll
  for (int s = 0; s < 16; ++s) {
#pragma unroll
    for (int v = 0; v < 2; ++v) {
      int n = 4 * s + khalf + v;            // k index = antenna n
      ar[s][v] = sMre[n * M_MEAS + mrow];   // A[m,k] = Mr[n,m]
      ai[s][v] = sMim[n * M_MEAS + mrow];
    }
  }

  const int acol = lane & 15;               // column within 16-wide a-tile
  for (int t = wave; t < D_A / 16; t += 8) {
    const int a0 = t * 16;
    v8f accR  = {0, 0, 0, 0, 0, 0, 0, 0};
    v8f accI1 = {0, 0, 0, 0, 0, 0, 0, 0};
    v8f accI2 = {0, 0, 0, 0, 0, 0, 0, 0};
#pragma unroll
    for (int s = 0; s < 16; ++s) {
      v2f br, bi;
#pragma unroll
      for (int v = 0; v < 2; ++v) {
        int n = 4 * s + khalf + v;          // B[k,n_col], mirrored A layout
        br[v] = Wre[n * D_A + a0 + acol];
        bi[v] = Wim[n * D_A + a0 + acol];
      }
      accR  = wmma_f32(ar[s], br, accR);    // Mr^T Wr
      accR  = wmma_f32(ai[s], bi, accR);    // + Mi^T Wi
      accI1 = wmma_f32(ar[s], bi, accI1);   // Mr^T Wi
      accI2 = wmma_f32(ai[s], br, accI2);   // Mi^T Wr  (im = I1 - I2)
    }
    // C/D layout: slot v holds row m = (lane>>4)*8 + v, col = lane&15.
    float partial = 0.f;
#pragma unroll
    for (int v = 0; v < 8; ++v) {
      float re = accR[v];
      float im = accI1[v] - accI2[v];
      partial += re * re + im * im;
    }
    partial += __shfl_xor(partial, 16, 32); // combine m=0..7 with m=8..15
    if (lane < 16)
      md2inv[(size_t)b * D_A + a0 + acol] = 1.0f / partial;
  }
}

// ---------------------------------------------------------------------------
// init: r = x, res = xm, active = 1, u = M[b] @ r[b]
// ---------------------------------------------------------------------------
__global__ void init_kernel(const float* __restrict__ xre, const float* __restrict__ xim,
                            const float* __restrict__ xmre, const float* __restrict__ xmim,
                            const float* __restrict__ Mre, const float* __restrict__ Mim,
                            float* __restrict__ r_re, float* __restrict__ r_im,
                            float* __restrict__ res_re, float* __restrict__ res_im,
                            float* __restrict__ u_re, float* __restrict__ u_im,
                            int* __restrict__ active) {
  const int b = blockIdx.x, n = threadIdx.x;  // 64 threads
  if (n < M_MEAS) {
    r_re[b * M_MEAS + n] = xre[b * M_MEAS + n];
    r_im[b * M_MEAS + n] = xim[b * M_MEAS + n];
  }
  res_re[b * N_ANT + n] = xmre[b * N_ANT + n];
  res_im[b * N_ANT + n] = xmim[b * N_ANT + n];
  if (n == 0) active[b] = 1;
  float ur = 0.f, ui = 0.f;
  for (int m = 0; m < M_MEAS; ++m) {
    float mr = Mre[(size_t)b * (N_ANT * M_MEAS) + n * M_MEAS + m];
    float mi = Mim[(size_t)b * (N_ANT * M_MEAS) + n * M_MEAS + m];
    float rr = xre[b * M_MEAS + m], ri = xim[b * M_MEAS + m];
    ur += mr * rr - mi * ri;
    ui += mr * ri + mi * rr;
  }
  u_re[b * N_ANT + n] = ur;
  u_im[b * N_ANT + n] = ui;
}

// ---------------------------------------------------------------------------
// select: V = U * conj(W); argmax_a |V[b,a]|^2 * md2inv[b,a].
//   V_re = Ur Wr + Ui Wi ; V_im = Ui Wr - Ur Wi
// One block per 16-row b-tile; 8 waves each own 32 contiguous a-tiles.
// ---------------------------------------------------------------------------
__global__ void __launch_bounds__(256)
select_kernel(const float* __restrict__ u_re, const float* __restrict__ u_im,
              const float* __restrict__ Wre, const float* __restrict__ Wim,
              const float* __restrict__ md2inv,
              int* __restrict__ idx_out,
              float* __restrict__ vre_out, float* __restrict__ vim_out) {
  const int b0 = blockIdx.x * 16;
  const int tid = threadIdx.x, lane = tid & 31, wave = tid >> 5;

  __shared__ float sUr[16 * N_ANT], sUi[16 * N_ANT];
  __shared__ float sMet[16][128];
  __shared__ int   sIdx[16][128];
  __shared__ float sVr[16][128], sVi[16][128];

  for (int i = tid; i < 16 * N_ANT; i += 256) {
    int bl = i >> 6, n = i & 63;
    sUr[i] = u_re[(b0 + bl) * N_ANT + n];
    sUi[i] = u_im[(b0 + bl) * N_ANT + n];
  }
  __syncthreads();

  const int brow  = lane & 15;               // A row = local b
  const int khalf = (lane >> 4) << 1;
  v2f ar[16], ai[16];
#pragma unroll
  for (int s = 0; s < 16; ++s) {
#pragma unroll
    for (int v = 0; v < 2; ++v) {
      int n = 4 * s + khalf + v;
      ar[s][v] = sUr[brow * N_ANT + n];
      ai[s][v] = sUi[brow * N_ANT + n];
    }
  }

  const int acol  = lane & 15;
  const int bhalf = (lane >> 4) << 3;        // C row base
  float bestM[8], bestR[8], bestI[8];
  int   bestA[8];
#pragma unroll
  for (int v = 0; v < 8; ++v) { bestM[v] = -1.f; bestA[v] = 0; bestR[v] = 0.f; bestI[v] = 0.f; }

  for (int t = wave * 32; t < wave * 32 + 32; ++t) {  // contiguous => a ascending
    const int a0 = t * 16;
    v8f accR  = {0, 0, 0, 0, 0, 0, 0, 0};
    v8f accI1 = {0, 0, 0, 0, 0, 0, 0, 0};
    v8f accI2 = {0, 0, 0, 0, 0, 0, 0, 0};
#pragma unroll
    for (int s = 0; s < 16; ++s) {
      v2f br, bi;
#pragma unroll
      for (int v = 0; v < 2; ++v) {
        int n = 4 * s + khalf + v;
        br[v] = Wre[n * D_A + a0 + acol];
        bi[v] = Wim[n * D_A + a0 + acol];
      }
      accR  = wmma_f32(ar[s], br, accR);    // Ur Wr
      accR  = wmma_f32(ai[s], bi, accR);    // + Ui Wi
      accI1 = wmma_f32(ai[s], br, accI1);   // Ui Wr
      accI2 = wmma_f32(ar[s], bi, accI2);   // Ur Wi  (im = I1 - I2)
    }
#pragma unroll
    for (int v = 0; v < 8; ++v) {
      float re = accR[v];
      float im = accI1[v] - accI2[v];
      int   a  = a0 + acol;
      float mi = md2inv[(size_t)(b0 + bhalf + v) * D_A + a];
      float met = (re * re + im * im) * mi;
      if (met > bestM[v]) { bestM[v] = met; bestA[v] = a; bestR[v] = re; bestI[v] = im; }
    }
  }
#pragma unroll
  for (int v = 0; v < 8; ++v) {
    int bl = bhalf + v, c = wave * 16 + acol;
    sMet[bl][c] = bestM[v]; sIdx[bl][c] = bestA[v];
    sVr[bl][c]  = bestR[v]; sVi[bl][c]  = bestI[v];
  }
  __syncthreads();
  if (tid < 16) {
    float bm = -1.f, br_ = 0.f, bi_ = 0.f;
    int   ba = D_A;
    for (int c = 0; c < 128; ++c) {
      float m = sMet[tid][c];
      int   a = sIdx[tid][c];
      if (m > bm || (m == bm && a < ba)) { bm = m; ba = a; br_ = sVr[tid][c]; bi_ = sVi[tid][c]; }
    }
    idx_out[b0 + tid] = ba;
    vre_out[b0 + tid] = br_;
    vim_out[b0 + tid] = bi_;
  }
}

// ---------------------------------------------------------------------------
// update: active gating, r -= val*col, res -= val*colE, refresh u.
// One block per b (64 threads).
// ---------------------------------------------------------------------------
__global__ void update_kernel(const float* __restrict__ Mre, const float* __restrict__ Mim,
                              const float* __restrict__ Wre, const float* __restrict__ Wim,
                              const float* __restrict__ sigma, const int* __restrict__ Lp,
                              const float* __restrict__ md2inv,
                              const float* __restrict__ enorm_inv,
                              const int* __restrict__ idx_in,
                              const float* __restrict__ vre_in, const float* __restrict__ vim_in,
                              float* __restrict__ r_re, float* __restrict__ r_im,
                              float* __restrict__ res_re, float* __restrict__ res_im,
                              float* __restrict__ u_re, float* __restrict__ u_im,
                              int* __restrict__ active) {
  const int b = blockIdx.x, tid = threadIdx.x;  // 64 threads
  __shared__ float srn[M_MEAS];
  __shared__ float sr_re[M_MEAS], sr_im[M_MEAS];
  __shared__ int s_act;

  if (tid < M_MEAS) {
    float rr = r_re[b * M_MEAS + tid], ri = r_im[b * M_MEAS + tid];
    srn[tid] = rr * rr + ri * ri;
    sr_re[tid] = rr; sr_im[tid] = ri;
  }
  __syncthreads();
  if (tid == 0) {
    float rn2 = 0.f;
    for (int m = 0; m < M_MEAS; ++m) rn2 += srn[m];
    float sg  = sigma[b];
    float thr = sg * sg * (float)N_ANT * (float)Lp[0];
    int act = active[b] && (rn2 >= thr);
    active[b] = act;
    s_act = act;
  }
  __syncthreads();
  if (!s_act) return;

  const int   t    = idx_in[b];
  const float mi   = md2inv[(size_t)b * D_A + t];
  const float invn = sqrtf(mi);                  // 1 / md_norm
  const float valr = vre_in[b] * invn;           // val = corr at winner
  const float vali = vim_in[b] * invn;

  if (tid < M_MEAS) {
    // col[m] = (sum_n conj(M[b,n,m]) W[n,t]) * invn
    float cr = 0.f, ci = 0.f;
    for (int n = 0; n < N_ANT; ++n) {
      float mr = Mre[(size_t)b * (N_ANT * M_MEAS) + n * M_MEAS + tid];
      float mm = Mim[(size_t)b * (N_ANT * M_MEAS) + n * M_MEAS + tid];
      float wr = Wre[n * D_A + t], wi = Wim[n * D_A + t];
      cr += mr * wr + mm * wi;
      ci += mr * wi - mm * wr;
    }
    float colr = cr * invn, coli = ci * invn;
    float dr = valr * colr - vali * coli;
    float di = valr * coli + vali * colr;
    float nr = sr_re[tid] - dr, ni = sr_im[tid] - di;
    r_re[b * M_MEAS + tid] = nr;
    r_im[b * M_MEAS + tid] = ni;
    sr_re[tid] = nr; sr_im[tid] = ni;
  }
  {
    float wr = Wre[tid * D_A + t], wi = Wim[tid * D_A + t];
    float en = enorm_inv[t];
    float cer = wr * en, cei = wi * en;
    res_re[b * N_ANT + tid] -= valr * cer - vali * cei;
    res_im[b * N_ANT + tid] -= valr * cei + vali * cer;
  }
  __syncthreads();
  // u[b,n] for next iteration from updated r
  float ur = 0.f, ui = 0.f;
  for (int m = 0; m < M_MEAS; ++m) {
    float mr = Mre[(size_t)b * (N_ANT * M_MEAS) + tid * M_MEAS + m];
    float mm = Mim[(size_t)b * (N_ANT * M_MEAS) + tid * M_MEAS + m];
    ur += mr * sr_re[m] - mm * sr_im[m];
    ui += mr * sr_im[m] + mm * sr_re[m];
  }
  u_re[b * N_ANT + tid] = ur;
  u_im[b * N_ANT + tid] = ui;
}

// ---------------------------------------------------------------------------
// finalize: (residual, x - residual, xm - res) interleaved complex64.
// ---------------------------------------------------------------------------
__global__ void finalize_kernel(const float* __restrict__ xre, const float* __restrict__ xim,
                                const float* __restrict__ xmre, const float* __restrict__ xmim,
                                const float* __restrict__ r_re, const float* __restrict__ r_im,
                                const float* __restrict__ res_re, const float* __restrict__ res_im,
                                float* __restrict__ out) {
  int i = blockIdx.x * blockDim.x + threadIdx.x;
  if (i < B_SZ * M_MEAS) {
    float rr = r_re[i], ri = r_im[i];
    out[2 * i]     = rr;
    out[2 * i + 1] = ri;
    out[2 * B_SZ * M_MEAS + 2 * i]     = xre[i] - rr;
    out[2 * B_SZ * M_MEAS + 2 * i + 1] = xim[i] - ri;
  }
  if (i < B_SZ * N_ANT) {
    out[4 * B_SZ * M_MEAS + 2 * i]     = xmre[i] - res_re[i];
    out[4 * B_SZ * M_MEAS + 2 * i + 1] = xmim[i] - res_im[i];
  }
}

// ---------------------------------------------------------------------------
extern "C" void kernel_launch(void* const* d_in, const int* in_sizes, int n_in,
                              void* d_out, int out_size, void* d_ws, size_t ws_size,
                              hipStream_t stream) {
  const float* xre  = (const float*)d_in[0];
  const float* xim  = (const float*)d_in[1];
  const float* xmre = (const float*)d_in[2];
  const float* xmim = (const float*)d_in[3];
  const float* Mre  = (const float*)d_in[4];
  const float* Mim  = (const float*)d_in[5];
  const float* Wre  = (const float*)d_in[6];
  const float* Wim  = (const float*)d_in[7];
  const float* sig  = (const float*)d_in[8];
  const int*   Lp   = (const int*)d_in[9];
  float* out = (float*)d_out;

  // workspace layout (float granularity)
  float* ws = (float*)d_ws;
  size_t off = 0;
  float* md2inv    = ws + off; off += (size_t)B_SZ * D_A;   // 16 MB
  float* u_re      = ws + off; off += (size_t)B_SZ * N_ANT;
  float* u_im      = ws + off; off += (size_t)B_SZ * N_ANT;
  float* r_re      = ws + off; off += (size_t)B_SZ * M_MEAS;
  float* r_im      = ws + off; off += (size_t)B_SZ * M_MEAS;
  float* res_re    = ws + off; off += (size_t)B_SZ * N_ANT;
  float* res_im    = ws + off; off += (size_t)B_SZ * N_ANT;
  float* enorm_inv = ws + off; off += (size_t)D_A;
  float* vre       = ws + off; off += (size_t)B_SZ;
  float* vim       = ws + off; off += (size_t)B_SZ;
  int*   idx       = (int*)(ws + off); off += (size_t)B_SZ;
  int*   active    = (int*)(ws + off); off += (size_t)B_SZ;

  enorm_kernel<<<(D_A + 255) / 256, 256, 0, stream>>>(Wre, Wim, enorm_inv);
  norms_kernel<<<B_SZ, 256, 0, stream>>>(Mre, Mim, Wre, Wim, md2inv);
  init_kernel<<<B_SZ, N_ANT, 0, stream>>>(xre, xim, xmre, xmim, Mre, Mim,
                                          r_re, r_im, res_re, res_im,
                                          u_re, u_im, active);
  for (int it = 0; it < ITERS; ++it) {
    select_kernel<<<B_SZ / 16, 256, 0, stream>>>(u_re, u_im, Wre, Wim, md2inv,
                                                 idx, vre, vim);
    update_kernel<<<B_SZ, N_ANT, 0, stream>>>(Mre, Mim, Wre, Wim, sig, Lp,
                                              md2inv, enorm_inv, idx, vre, vim,
                                              r_re, r_im, res_re, res_im,
                                              u_re, u_im, active);
  }
  int ntot = B_SZ * N_ANT;
  finalize_kernel<<<(ntot + 255) / 256, 256, 0, stream>>>(
      xre, xim, xmre, xmim, r_re, r_im, res_re, res_im, out);
}